// CustomMultiheadAttention_38963943309893
// MI455X (gfx1250) — compile-verified
//
#include <hip/hip_runtime.h>

// Complex multi-head attention for MI455X (gfx1250), fp32 path via
// V_WMMA_F32_16X16X4_F32. All matrices stay complex64 (interleaved float2).
// Inner loops are software-pipelined (register double-buffered) so
// global_load_b128 latency overlaps the WMMA block instead of stalling on
// s_wait_loadcnt 0 right after each load.

typedef __attribute__((ext_vector_type(2))) float v2f;
typedef __attribute__((ext_vector_type(8))) float v8f;

constexpr int kB = 4, kT = 1024, kC = 512, kH = 8, kD = 64;
constexpr float kScale = 0.125f; // D^-0.5 = 64^-0.5

// One complex 16x16x4 MAC step = 4 real WMMAs.
//   accr += Ar*Br - Ai*Bi ; acci += Ar*Bi + Ai*Br
// Ai is negated in VALU (f32 WMMA NEG bits are C-negate only per ISA).
__device__ __forceinline__ void cwmma(v8f& accr, v8f& acci,
                                      v2f ar, v2f ai, v2f br, v2f bi) {
  v2f ain = -ai;
  accr = __builtin_amdgcn_wmma_f32_16x16x4_f32(false, ar,  false, br,
                                               (short)0, accr, false, false);
  accr = __builtin_amdgcn_wmma_f32_16x16x4_f32(false, ain, false, bi,
                                               (short)0, accr, false, false);
  acci = __builtin_amdgcn_wmma_f32_16x16x4_f32(false, ar,  false, bi,
                                               (short)0, acci, false, false);
  acci = __builtin_amdgcn_wmma_f32_16x16x4_f32(false, ai,  false, br,
                                               (short)0, acci, false, false);
}

// Run the 16-WMMA block for one k-step from raw interleaved fragments.
__device__ __forceinline__ void cstep(v8f accr[4], v8f acci[4],
                                      const float4& a, const float4 b[4]) {
  const v2f ar = {a.x, a.z}, ai = {a.y, a.w};
#pragma unroll
  for (int t = 0; t < 4; ++t)
    cwmma(accr[t], acci[t], ar, ai, v2f{b[t].x, b[t].z}, v2f{b[t].y, b[t].w});
}

// ---------------------------------------------------------------------------
// Complex GEMM-NT with bias:  out[m,n] = sum_k A[m,k] * W[n,k] + bias[n]
// Wave tile: 16(M) x 64(N). Block = 8 waves covering 128(M) x 64(N).
// bhtd==1 scatters output from (B*T, H*D) row-major to (B,H,T,D).
// ---------------------------------------------------------------------------
__global__ __launch_bounds__(256) void cgemm_nt_proj(
    const float2* __restrict__ A, const float2* __restrict__ W,
    const float2* __restrict__ bias, float2* __restrict__ out,
    int M, int N, int K, int bhtd) {
  const int lane = threadIdx.x & 31;
  const int wave = threadIdx.x >> 5;
  const int m0 = (blockIdx.x * 8 + wave) * 16;
  const int n0 = blockIdx.y * 64;
  const int r  = lane & 15;          // M row (A) / N col (B) inside tile
  const int kb = (lane >> 4) << 1;   // K sub-offset {0,2}

  v8f zero{};
  v8f accr[4] = {zero, zero, zero, zero};
  v8f acci[4] = {zero, zero, zero, zero};

  const float4* Ap = reinterpret_cast<const float4*>(A + (size_t)(m0 + r) * K);
  const float4* Wp[4];
#pragma unroll
  for (int t = 0; t < 4; ++t)
    Wp[t] = reinterpret_cast<const float4*>(W + (size_t)(n0 + t * 16 + r) * K);

  // Software-pipelined k loop: prefetch next fragments, compute on current.
  const int idx0 = kb >> 1;
  float4 a_cur = Ap[idx0];
  float4 b_cur[4];
#pragma unroll
  for (int t = 0; t < 4; ++t) b_cur[t] = Wp[t][idx0];

  for (int k = 4; k < K; k += 4) {
    const int idx = (k + kb) >> 1;   // float4 = 2 consecutive complex along K
    const float4 a_nxt = Ap[idx];
    float4 b_nxt[4];
#pragma unroll
    for (int t = 0; t < 4; ++t) b_nxt[t] = Wp[t][idx];

    cstep(accr, acci, a_cur, b_cur);

    a_cur = a_nxt;
#pragma unroll
    for (int t = 0; t < 4; ++t) b_cur[t] = b_nxt[t];
  }
  cstep(accr, acci, a_cur, b_cur);

  const int mo = (lane >> 4) * 8;    // C/D layout: VGPR j -> M = j + 8*(lane>>4)
#pragma unroll
  for (int t = 0; t < 4; ++t) {
    const int col = n0 + t * 16 + r;
    const float2 bv = bias[col];
#pragma unroll
    for (int j = 0; j < 8; ++j) {
      const int m = m0 + mo + j;
      const float2 o = make_float2(accr[t][j] + bv.x, acci[t][j] + bv.y);
      size_t off;
      if (bhtd) {
        const int bb = m >> 10, tt = m & (kT - 1);
        const int hh = col >> 6, dd = col & (kD - 1);
        off = (((size_t)(bb * kH + hh) * kT) + tt) * kD + dd;
      } else {
        off = (size_t)m * N + col;
      }
      out[off] = o;
    }
  }
}

// ---------------------------------------------------------------------------
// attn_weights[bh] = crelu( (Q[bh] @ K[bh]^T) * scale ),  M=N=T, K=D
// ---------------------------------------------------------------------------
__global__ __launch_bounds__(256) void attn_qk_kernel(
    const float2* __restrict__ Q, const float2* __restrict__ Kmat,
    float2* __restrict__ outw) {
  const int bh = blockIdx.z;
  const float2* Qb = Q + (size_t)bh * kT * kD;
  const float2* Kb = Kmat + (size_t)bh * kT * kD;
  float2* Ob = outw + (size_t)bh * kT * kT;

  const int lane = threadIdx.x & 31;
  const int wave = threadIdx.x >> 5;
  const int m0 = (blockIdx.x * 8 + wave) * 16;
  const int n0 = blockIdx.y * 64;
  const int r  = lane & 15;
  const int kb = (lane >> 4) << 1;

  v8f zero{};
  v8f accr[4] = {zero, zero, zero, zero};
  v8f acci[4] = {zero, zero, zero, zero};

  const float4* Ap = reinterpret_cast<const float4*>(Qb + (size_t)(m0 + r) * kD);
  const float4* Bp[4];
#pragma unroll
  for (int t = 0; t < 4; ++t)
    Bp[t] = reinterpret_cast<const float4*>(Kb + (size_t)(n0 + t * 16 + r) * kD);

  const int idx0 = kb >> 1;
  float4 a_cur = Ap[idx0];
  float4 b_cur[4];
#pragma unroll
  for (int t = 0; t < 4; ++t) b_cur[t] = Bp[t][idx0];

#pragma unroll
  for (int k = 4; k < kD; k += 4) {
    const int idx = (k + kb) >> 1;
    const float4 a_nxt = Ap[idx];
    float4 b_nxt[4];
#pragma unroll
    for (int t = 0; t < 4; ++t) b_nxt[t] = Bp[t][idx];

    cstep(accr, acci, a_cur, b_cur);

    a_cur = a_nxt;
#pragma unroll
    for (int t = 0; t < 4; ++t) b_cur[t] = b_nxt[t];
  }
  cstep(accr, acci, a_cur, b_cur);

  const int mo = (lane >> 4) * 8;
#pragma unroll
  for (int t = 0; t < 4; ++t) {
    const int col = n0 + t * 16 + r;
#pragma unroll
    for (int j = 0; j < 8; ++j) {
      const int m = m0 + mo + j;
      const float re = fmaxf(accr[t][j] * kScale, 0.0f);
      const float im = fmaxf(acci[t][j] * kScale, 0.0f);
      Ob[(size_t)m * kT + col] = make_float2(re, im);
    }
  }
}

// ---------------------------------------------------------------------------
// attn_out[bh] = attn_weights[bh] @ V[bh]   (GEMM-NN: M=T, N=D, K=T)
// Output scattered to (B, T, H*D) so the final projection reads it row-major.
// ---------------------------------------------------------------------------
__global__ __launch_bounds__(256) void attn_av_kernel(
    const float2* __restrict__ Wt, const float2* __restrict__ V,
    float2* __restrict__ out) {
  const int bh = blockIdx.z;
  const int bb = bh >> 3, hh = bh & 7;
  const float2* Ab = Wt + (size_t)bh * kT * kT;
  const float2* Vb = V + (size_t)bh * kT * kD;

  const int lane = threadIdx.x & 31;
  const int wave = threadIdx.x >> 5;
  const int m0 = (blockIdx.x * 8 + wave) * 16;
  const int r  = lane & 15;
  const int kb = (lane >> 4) << 1;

  v8f zero{};
  v8f accr[4] = {zero, zero, zero, zero};
  v8f acci[4] = {zero, zero, zero, zero};

  const float4* Ap = reinterpret_cast<const float4*>(Ab + (size_t)(m0 + r) * kT);

  // Pipeline: a as float4, B operand as two float2 rows per tile.
  float4 a_cur = Ap[kb >> 1];
  float2 b0_cur[4], b1_cur[4];
#pragma unroll
  for (int t = 0; t < 4; ++t) {
    const int col = t * 16 + r;
    b0_cur[t] = Vb[(size_t)kb * kD + col];
    b1_cur[t] = Vb[(size_t)(kb + 1) * kD + col];
  }

  for (int k = 4; k < kT; k += 4) {
    const int krow = k + kb;
    const float4 a_nxt = Ap[krow >> 1];
    float2 b0_nxt[4], b1_nxt[4];
#pragma unroll
    for (int t = 0; t < 4; ++t) {
      const int col = t * 16 + r;
      b0_nxt[t] = Vb[(size_t)krow * kD + col];
      b1_nxt[t] = Vb[(size_t)(krow + 1) * kD + col];
    }

    {
      const v2f ar = {a_cur.x, a_cur.z}, ai = {a_cur.y, a_cur.w};
#pragma unroll
      for (int t = 0; t < 4; ++t)
        cwmma(accr[t], acci[t], ar, ai,
              v2f{b0_cur[t].x, b1_cur[t].x}, v2f{b0_cur[t].y, b1_cur[t].y});
    }

    a_cur = a_nxt;
#pragma unroll
    for (int t = 0; t < 4; ++t) { b0_cur[t] = b0_nxt[t]; b1_cur[t] = b1_nxt[t]; }
  }
  {
    const v2f ar = {a_cur.x, a_cur.z}, ai = {a_cur.y, a_cur.w};
#pragma unroll
    for (int t = 0; t < 4; ++t)
      cwmma(accr[t], acci[t], ar, ai,
            v2f{b0_cur[t].x, b1_cur[t].x}, v2f{b0_cur[t].y, b1_cur[t].y});
  }

  const int mo = (lane >> 4) * 8;
#pragma unroll
  for (int t = 0; t < 4; ++t) {
    const int col = t * 16 + r;
#pragma unroll
    for (int j = 0; j < 8; ++j) {
      const int m = m0 + mo + j;
      out[((size_t)(bb * kT + m)) * kC + hh * kD + col] =
          make_float2(accr[t][j], acci[t][j]);
    }
  }
}

// ---------------------------------------------------------------------------
extern "C" void kernel_launch(void* const* d_in, const int* in_sizes, int n_in,
                              void* d_out, int out_size, void* d_ws,
                              size_t ws_size, hipStream_t stream) {
  (void)in_sizes; (void)n_in; (void)out_size; (void)ws_size;

  const float2* query = (const float2*)d_in[0];
  const float2* Wq = (const float2*)d_in[1];
  const float2* bq = (const float2*)d_in[2];
  const float2* Wk = (const float2*)d_in[3];
  const float2* bk = (const float2*)d_in[4];
  const float2* Wv = (const float2*)d_in[5];
  const float2* bv = (const float2*)d_in[6];
  const float2* Wo = (const float2*)d_in[7];
  const float2* bo = (const float2*)d_in[8];

  const size_t nBTC = (size_t)kB * kT * kC;      // 2,097,152 complex
  float2* out_attn = (float2*)d_out;             // (B,T,C) complex
  float2* out_w    = out_attn + nBTC;            // (B,H,T,T) complex

  float2* Qb  = (float2*)d_ws;                   // (B,H,T,D)
  float2* Kb  = Qb + nBTC;
  float2* Vb  = Kb + nBTC;
  float2* Tmp = Vb + nBTC;                       // (B,T,C) pre-projection

  const dim3 blk(256);
  const dim3 gproj(kB * kT / 128, kC / 64);      // (32, 8)

  cgemm_nt_proj<<<gproj, blk, 0, stream>>>(query, Wq, bq, Qb,
                                           kB * kT, kC, kC, 1);
  cgemm_nt_proj<<<gproj, blk, 0, stream>>>(query, Wk, bk, Kb,
                                           kB * kT, kC, kC, 1);
  cgemm_nt_proj<<<gproj, blk, 0, stream>>>(query, Wv, bv, Vb,
                                           kB * kT, kC, kC, 1);

  attn_qk_kernel<<<dim3(kT / 128, kT / 64, kB * kH), blk, 0, stream>>>(
      Qb, Kb, out_w);

  attn_av_kernel<<<dim3(kT / 128, 1, kB * kH), blk, 0, stream>>>(
      out_w, Vb, Tmp);

  cgemm_nt_proj<<<gproj, blk, 0, stream>>>(Tmp, Wo, bo, out_attn,
                                           kB * kT, kC, kC, 0);
}